// BilateralSliceApply_76433237999691
// MI455X (gfx1250) — compile-verified
//
#include <hip/hip_runtime.h>
#include <hip/hip_bf16.h>
#include <cstdint>

// Problem constants (fixed by reference setup_inputs)
#define NB   4
#define GH   16
#define GW   16
#define GD   8
#define HH   2048
#define WW   2048

constexpr int HW         = HH * WW;        // 4,194,304 pixels per image
constexpr int CELLS      = GH * GW * GD;   // 2048 grid cells per image
constexpr int GRID_ELEMS = 2 * CELLS;      // 4096 floats (a-plane + b-plane)

typedef float v4f __attribute__((ext_vector_type(4)));   // native vector for NT builtins

__device__ __forceinline__ int iclamp(int v, int lo, int hi) {
    return v < lo ? lo : (v > hi ? hi : v);
}

// grid.x = 512: h0 = blockIdx.x>>1 (0..255, wave-uniform), column-half = blockIdx.x&1.
// Each thread owns a fixed 4-pixel column group and walks 8 rows (h0 + 256*it).
__global__ __launch_bounds__(256) void bsa_kernel(
    const float* __restrict__ bgrid,   // [N, 2, GH, GW, GD]
    const float* __restrict__ guide,   // [N, H, W]
    const float* __restrict__ input,   // [N, 3, H, W]
    float* __restrict__ out)           // [N, 1, H, W]
{
    __shared__ float2 s_grid[CELLS];   // interleaved (a,b) per cell, 16 KB

    const int n = blockIdx.y;

    // ---- Stage grid slice to LDS via CDNA5 async global->LDS DMA ----
    // Global layout [c][cell] -> LDS interleaved float2[cell]; per-lane
    // addresses on both sides let the async unit do the channel interleave.
    {
        const float*   gsrc     = bgrid + (size_t)n * GRID_ELEMS;
        const uint32_t lds_base = (uint32_t)(uintptr_t)(void*)&s_grid[0];
        #pragma unroll
        for (int e = (int)threadIdx.x; e < GRID_ELEMS; e += 256) {
            uint64_t ga = (uint64_t)(uintptr_t)(gsrc + e);
            uint32_t la = lds_base
                        + ((uint32_t)(e & (CELLS - 1)) << 3)   // cell*8
                        + ((uint32_t)(e >> 11) << 2);          // c*4
            asm volatile("global_load_async_to_lds_b32 %0, %1, off"
                         :: "v"(la), "v"(ga) : "memory");
        }
        asm volatile("s_wait_asynccnt 0" ::: "memory");
    }
    __syncthreads();

    const float SXY = (float)(GW - 1) / (float)(WW - 1);   // 15/2047
    const float SZ  = 0.5f * (float)(GD - 1);              // 3.5

    const int h0 = (int)(blockIdx.x >> 1);                              // uniform
    const int w  = ((((int)blockIdx.x & 1) << 8) + (int)threadIdx.x) << 2;

    // ---- Per-thread fixed x-interpolation (hoisted out of the row loop) ----
    int   xo0[4], xo1[4];        // element offsets x*GD
    float wx0[4], wx1[4];
    #pragma unroll
    for (int k = 0; k < 4; ++k) {
        const float fx = (float)(w + k) * SXY;
        const int   xf = (int)floorf(fx);
        const float tx = fx - (float)xf;
        const int   x0 = iclamp(xf, 0, GW - 1);
        xo0[k] = x0 * GD;
        xo1[k] = iclamp(x0 + 1, 0, GW - 1) * GD;
        wx1[k] = tx;
        wx0[k] = 1.0f - tx;
    }

    const float* gplane = guide + (size_t)n * HW;
    const float* rp     = input + (size_t)n * 3 * HW;
    const float* gp     = rp + HW;
    const float* bp     = gp + HW;
    float*       op     = out + (size_t)n * HW;

    #pragma unroll
    for (int it = 0; it < 8; ++it) {
        const int h = h0 + (it << 8);
        const int p = (h << 11) + w;

        // y interpolation (wave-uniform)
        const float fy  = (float)h * SXY;
        const int   yf  = (int)floorf(fy);
        const float ty  = fy - (float)yf;
        const int   y0  = iclamp(yf, 0, GH - 1);
        const int   y1  = iclamp(y0 + 1, 0, GH - 1);
        const int   yb0 = y0 * (GW * GD);
        const int   yb1 = y1 * (GW * GD);
        const float wy0 = 1.0f - ty, wy1 = ty;

        // 128-bit non-temporal streaming loads (zero reuse -> keep L2 clean)
        const v4f gv4 = __builtin_nontemporal_load(reinterpret_cast<const v4f*>(gplane + p));
        const v4f rv4 = __builtin_nontemporal_load(reinterpret_cast<const v4f*>(rp + p));
        const v4f gg4 = __builtin_nontemporal_load(reinterpret_cast<const v4f*>(gp + p));
        const v4f bv4 = __builtin_nontemporal_load(reinterpret_cast<const v4f*>(bp + p));

        v4f o4;
        #pragma unroll
        for (int k = 0; k < 4; ++k) {
            // z interpolation from guide
            const float gz  = fmaf(gv4[k], SZ, SZ);       // (g+1)*3.5
            const float zff = floorf(gz);
            const float tz  = gz - zff;
            const int   zf  = (int)zff;
            const int   z0  = iclamp(zf, 0, GD - 1);
            const int   z1  = iclamp(z0 + 1, 0, GD - 1);
            const float wz0 = 1.0f - tz, wz1 = tz;

            // bilinear corner weights (wy uniform, wx thread-fixed)
            const float w00 = wy0 * wx0[k], w01 = wy0 * wx1[k];
            const float w10 = wy1 * wx0[k], w11 = wy1 * wx1[k];

            float a = 0.0f, b = 0.0f;
            auto acc = [&](int base, float wxy) {
                const float2 c0 = s_grid[base + z0];   // ds_load_b64 (a,b)@z0
                const float2 c1 = s_grid[base + z1];   // ds_load_b64 (a,b)@z1
                a = fmaf(wxy, fmaf(wz0, c0.x, wz1 * c1.x), a);
                b = fmaf(wxy, fmaf(wz0, c0.y, wz1 * c1.y), b);
            };
            acc(yb0 + xo0[k], w00);
            acc(yb0 + xo1[k], w01);
            acc(yb1 + xo0[k], w10);
            acc(yb1 + xo1[k], w11);

            // ApplyCoeffs: out = a*(r+g+b) + b
            o4[k] = fmaf(a, rv4[k] + gg4[k] + bv4[k], b);
        }

        __builtin_nontemporal_store(o4, reinterpret_cast<v4f*>(op + p));
    }
}

extern "C" void kernel_launch(void* const* d_in, const int* in_sizes, int n_in,
                              void* d_out, int out_size, void* d_ws, size_t ws_size,
                              hipStream_t stream) {
    const float* bgrid = (const float*)d_in[0];  // [4,2,16,16,8]
    const float* guide = (const float*)d_in[1];  // [4,2048,2048]
    const float* input = (const float*)d_in[2];  // [4,3,2048,2048]
    float*       out   = (float*)d_out;          // [4,1,2048,2048]

    dim3 grid(512, NB);
    dim3 block(256);
    hipLaunchKernelGGL(bsa_kernel, grid, block, 0, stream, bgrid, guide, input, out);
}